// BipartiteGraphConvolution_25993142075503
// MI455X (gfx1250) — compile-verified
//
#include <hip/hip_runtime.h>

#define N_CELLS 100000
#define N_GENES 20000
#define N_EDGES 640000
#define F 128
#define NHALF 64            // stage half of the N dimension per pass (64 cols)

typedef __attribute__((ext_vector_type(2))) float v2f;
typedef __attribute__((ext_vector_type(8))) float v8f;

// ---------------- zero workspace (neighbor accumulators) ----------------
__global__ __launch_bounds__(256) void zero_kernel(float4* __restrict__ ws, int n4) {
    int i = blockIdx.x * blockDim.x + threadIdx.x;
    if (i < n4) ws[i] = float4{0.f, 0.f, 0.f, 0.f};
}

// ---------------- edge scatter: one wave32 per edge ----------------
// lane handles features [lane*4, lane*4+4)
__global__ __launch_bounds__(256) void scatter_kernel(
    const float* __restrict__ cell_x, const float* __restrict__ gene_x,
    const int* __restrict__ erow, const int* __restrict__ ecol,
    const float* __restrict__ eval,
    float* __restrict__ cellN, float* __restrict__ geneN) {
    int e = (blockIdx.x * blockDim.x + threadIdx.x) >> 5;
    if (e >= N_EDGES) return;
    int lane = threadIdx.x & 31;
    int r = erow[e];
    int c = ecol[e];
    float v = eval[e];
    int f = lane << 2;

    // cellN[r] += v * gene_x[c]
    float4 g = *(const float4*)(gene_x + (size_t)c * F + f);
    float* cd = cellN + (size_t)r * F + f;
    __hip_atomic_fetch_add(cd + 0, v * g.x, __ATOMIC_RELAXED, __HIP_MEMORY_SCOPE_AGENT);
    __hip_atomic_fetch_add(cd + 1, v * g.y, __ATOMIC_RELAXED, __HIP_MEMORY_SCOPE_AGENT);
    __hip_atomic_fetch_add(cd + 2, v * g.z, __ATOMIC_RELAXED, __HIP_MEMORY_SCOPE_AGENT);
    __hip_atomic_fetch_add(cd + 3, v * g.w, __ATOMIC_RELAXED, __HIP_MEMORY_SCOPE_AGENT);

    // geneN[c] += v * cell_x[r]
    float4 cx = *(const float4*)(cell_x + (size_t)r * F + f);
    float* gd = geneN + (size_t)c * F + f;
    __hip_atomic_fetch_add(gd + 0, v * cx.x, __ATOMIC_RELAXED, __HIP_MEMORY_SCOPE_AGENT);
    __hip_atomic_fetch_add(gd + 1, v * cx.y, __ATOMIC_RELAXED, __HIP_MEMORY_SCOPE_AGENT);
    __hip_atomic_fetch_add(gd + 2, v * cx.z, __ATOMIC_RELAXED, __HIP_MEMORY_SCOPE_AGENT);
    __hip_atomic_fetch_add(gd + 3, v * cx.w, __ATOMIC_RELAXED, __HIP_MEMORY_SCOPE_AGENT);
}

// ---------------- fused dual-GEMM + bias + relu, WMMA f32 16x16x4 ----------------
// out = relu(X@W1 + N@W2 + (b1+b2)), one side (cell or gene) per launch.
// Block = 8 wave32s; each wave owns one 16-row M tile. Weights are staged in
// LDS half-N at a time (2 x 128x64 f32 = 64 KB) and shared by all 8 waves.
__global__ __launch_bounds__(256) void gemm_bias_relu_kernel(
    const float* __restrict__ X,  const float* __restrict__ Nm,
    const float* __restrict__ W1, const float* __restrict__ B1,
    const float* __restrict__ W2, const float* __restrict__ B2,
    float* __restrict__ out, int nTiles) {
    __shared__ float sW[2][F * NHALF];   // [matrix][k * NHALF + nn], 64 KB

    int tid    = threadIdx.x;
    int waveId = tid >> 5;
    int lane   = tid & 31;
    int tile   = blockIdx.x * 8 + waveId;          // wave-uniform
    bool active = (tile < nTiles);

    int lhalf = lane >> 4;   // K-pair select: {0,1} vs {2,3}
    int l15   = lane & 15;   // A row / B,C column within 16-tile
    int m0    = tile * 16;
    const float* Xrow = X  + (size_t)(m0 + l15) * F + lhalf * 2;
    const float* Nrow = Nm + (size_t)(m0 + l15) * F + lhalf * 2;

    for (int npass = 0; npass < 2; ++npass) {
        int nbase = npass * NHALF;

        // ---- cooperative stage: W1/W2 cols [nbase, nbase+64) into LDS ----
        // 2 matrices x 128 rows x 16 float4 = 4096 float4, 256 threads -> 16 ea.
        for (int i = tid; i < 2 * F * (NHALF / 4); i += 256) {
            int m  = i >> 11;               // 0 -> W1, 1 -> W2  (2048 float4 each)
            int j  = i & 2047;
            int k  = j >> 4;                // row 0..127
            int c4 = j & 15;                // float4 within 64-col half
            const float* src = (m == 0 ? W1 : W2) + (size_t)k * F + nbase + c4 * 4;
            float4 w = *(const float4*)src;
            *(float4*)(&sW[m][k * NHALF + c4 * 4]) = w;
        }
        __syncthreads();

        if (active) {
            for (int t = 0; t < NHALF / 16; ++t) {       // 4 col-tiles per pass
                int nn  = t * 16 + l15;                  // LDS column
                int col = nbase + nn;                    // global column
                v8f acc = {};
#pragma unroll 8
                for (int k0 = 0; k0 < F; k0 += 4) {
                    int kb = k0 + lhalf * 2;
                    // A = X[16x4] (global), B = W1[4x16] (LDS)
                    v2f a;  a.x  = Xrow[k0];  a.y  = Xrow[k0 + 1];
                    v2f b;  b.x  = sW[0][kb * NHALF + nn];
                            b.y  = sW[0][(kb + 1) * NHALF + nn];
                    acc = __builtin_amdgcn_wmma_f32_16x16x4_f32(
                              false, a, false, b, (short)0, acc, false, false);
                    // A = N[16x4] (global), B = W2[4x16] (LDS)
                    v2f a2; a2.x = Nrow[k0];  a2.y = Nrow[k0 + 1];
                    v2f b2; b2.x = sW[1][kb * NHALF + nn];
                            b2.y = sW[1][(kb + 1) * NHALF + nn];
                    acc = __builtin_amdgcn_wmma_f32_16x16x4_f32(
                              false, a2, false, b2, (short)0, acc, false, false);
                }
                float bias  = B1[col] + B2[col];
                int   rbase = m0 + lhalf * 8;
#pragma unroll
                for (int vv = 0; vv < 8; ++vv) {
                    float r = acc[vv] + bias;
                    out[(size_t)(rbase + vv) * F + col] = r > 0.f ? r : 0.f;
                }
            }
        }
        __syncthreads();   // safe to restage LDS for next pass
    }
}

extern "C" void kernel_launch(void* const* d_in, const int* in_sizes, int n_in,
                              void* d_out, int out_size, void* d_ws, size_t ws_size,
                              hipStream_t stream) {
    const float* cell_x = (const float*)d_in[0];
    const float* gene_x = (const float*)d_in[1];
    const int*   erow   = (const int*)d_in[2];
    const int*   ecol   = (const int*)d_in[3];
    const float* eval   = (const float*)d_in[4];
    const float* Wcs = (const float*)d_in[5];
    const float* bcs = (const float*)d_in[6];
    const float* Wcn = (const float*)d_in[7];
    const float* bcn = (const float*)d_in[8];
    const float* Wgs = (const float*)d_in[9];
    const float* bgs = (const float*)d_in[10];
    const float* Wgn = (const float*)d_in[11];
    const float* bgn = (const float*)d_in[12];

    float* cell_out = (float*)d_out;
    float* gene_out = cell_out + (size_t)N_CELLS * F;

    float* cellN = (float*)d_ws;
    float* geneN = cellN + (size_t)N_CELLS * F;

    // 1) zero neighbor accumulators (deterministic every call)
    int n4 = (N_CELLS + N_GENES) * F / 4;
    zero_kernel<<<(n4 + 255) / 256, 256, 0, stream>>>((float4*)d_ws, n4);

    // 2) edge scatter: one wave per edge -> 8 edges per 256-thread block
    long scatter_threads = (long)N_EDGES * 32;
    scatter_kernel<<<(int)((scatter_threads + 255) / 256), 256, 0, stream>>>(
        cell_x, gene_x, erow, ecol, eval, cellN, geneN);

    // 3) fused dual-GEMM + bias + relu, LDS-staged weights, one launch per side
    int cellTiles = N_CELLS / 16;   // 6250
    int geneTiles = N_GENES / 16;   // 1250
    gemm_bias_relu_kernel<<<(cellTiles + 7) / 8, 256, 0, stream>>>(
        cell_x, cellN, Wcs, bcs, Wcn, bcn, cell_out, cellTiles);
    gemm_bias_relu_kernel<<<(geneTiles + 7) / 8, 256, 0, stream>>>(
        gene_x, geneN, Wgs, bgs, Wgn, bgn, gene_out, geneTiles);
}